// DotProductAttHead_14972255994660
// MI455X (gfx1250) — compile-verified
//
#include <hip/hip_runtime.h>
#include <hip/hip_bf16.h>

#define NNODES 8192
#define DIM    256
#define NH     64
#define DEG    32
#define NE     (NNODES * DEG)
#define WSROW  192   // per-node workspace row: Q[0:64) K[64:128) V[128:192)

typedef float v2f __attribute__((ext_vector_type(2)));
typedef float v8f __attribute__((ext_vector_type(8)));

// ---------------------------------------------------------------------------
// Phase 1: fused QKV projection, full fp32 via V_WMMA_F32_16X16X4_F32.
// Each wave owns one (16-node x 16-channel) slot and produces the Q, K and V
// 16x16 tiles simultaneously: the A matrix (x tile) is loaded ONCE per k-step
// and feeds three WMMAs (3 accumulators). 512 M-tiles x 4 N-tiles = 2048 waves.
// ---------------------------------------------------------------------------
__global__ __launch_bounds__(256) void qkv_wmma_kernel(
    const float* __restrict__ x,
    const float* __restrict__ Wq, const float* __restrict__ bq,
    const float* __restrict__ Wk, const float* __restrict__ bk,
    const float* __restrict__ Wv,
    float* __restrict__ ws)
{
    const int lane = threadIdx.x & 31;
    const int wave = threadIdx.x >> 5;
    const int tile = blockIdx.x * 8 + wave;   // 2048 tiles total
    const int tm   = tile >> 2;               // M tile (16 nodes)
    const int tn   = tile & 3;                // N tile (16 of 64 channels)

    const int half = lane >> 4;               // 0: lanes 0-15, 1: lanes 16-31
    const int l15  = lane & 15;
    const int koff = half * 2;                // K sub-offset per lane half

    const int mrow = tm * 16 + l15;           // A-matrix row held by this lane
    const int ncol = tn * 16 + l15;           // B-matrix col held by this lane

    const float* __restrict__ xrow  = x  + (size_t)mrow * DIM;
    const float* __restrict__ wqrow = Wq + (size_t)ncol * DIM;
    const float* __restrict__ wkrow = Wk + (size_t)ncol * DIM;
    const float* __restrict__ wvrow = Wv + (size_t)ncol * DIM;

    v8f cq = {}, ck = {}, cv = {};
    #pragma unroll 4
    for (int k = 0; k < DIM; k += 4) {
        const v2f a  = *(const v2f*)(xrow  + k + koff);  // A 16x4 fp32 slice
        const v2f b0 = *(const v2f*)(wqrow + k + koff);  // B 4x16 slices
        const v2f b1 = *(const v2f*)(wkrow + k + koff);
        const v2f b2 = *(const v2f*)(wvrow + k + koff);
        cq = __builtin_amdgcn_wmma_f32_16x16x4_f32(false, a, false, b0,
                                                   (short)0, cq, false, false);
        ck = __builtin_amdgcn_wmma_f32_16x16x4_f32(false, a, false, b1,
                                                   (short)0, ck, false, false);
        cv = __builtin_amdgcn_wmma_f32_16x16x4_f32(false, a, false, b2,
                                                   (short)0, cv, false, false);
    }

    const float biasq = bq[ncol];
    const float biask = bk[ncol];

    #pragma unroll
    for (int r = 0; r < 8; ++r) {
        const int m = tm * 16 + half * 8 + r;       // D-matrix row for vgpr r
        float* __restrict__ row = ws + (size_t)m * WSROW + ncol;
        row[0]      = cq[r] + biasq;                // Q
        row[NH]     = ck[r] + biask;                // K
        row[2 * NH] = cv[r];                        // V (no bias)
    }
}

// ---------------------------------------------------------------------------
// Phase 2: per-node 32-neighbor attention. One wave32 per node:
// lane j handles neighbor j (score + softmax via shfl_xor butterflies),
// then p/neighbor-ids go through LDS and lane h accumulates channels h, h+32.
// ---------------------------------------------------------------------------
__global__ __launch_bounds__(256) void edge_attn_kernel(
    const float* __restrict__ ws,
    const int*   __restrict__ edge_dst,
    const int*   __restrict__ edge_type,
    const float* __restrict__ ek_table,
    float*       __restrict__ out)
{
    __shared__ float s_p[8][DEG];
    __shared__ int   s_nbr[8][DEG];

    const int lane = threadIdx.x & 31;
    const int wave = threadIdx.x >> 5;
    const int node = blockIdx.x * 8 + wave;

    const int   e   = node * DEG + lane;
    const int   nbr = edge_dst[e];
    const float ek  = ek_table[edge_type[e]];

    // q_i . k_nbr  (16B-aligned vector loads; q is lane-uniform -> broadcast)
    const float4* __restrict__ q = (const float4*)(ws + (size_t)node * WSROW);
    const float4* __restrict__ k = (const float4*)(ws + (size_t)nbr  * WSROW + NH);
    float s = 0.0f;
    #pragma unroll
    for (int h = 0; h < NH / 4; ++h) {
        const float4 qa = q[h], ka = k[h];
        s += qa.x * ka.x + qa.y * ka.y + qa.z * ka.z + qa.w * ka.w;
    }
    // ((q.k + ek)/H) / sqrt(H) = (q.k + ek) / 512
    s = (s + ek) * (1.0f / 512.0f);

    // wave32 softmax over the 32 neighbor scores
    float m = s;
    for (int off = 16; off > 0; off >>= 1)
        m = fmaxf(m, __shfl_xor(m, off, 32));
    const float ex = __expf(s - m);
    float sum = ex;
    for (int off = 16; off > 0; off >>= 1)
        sum += __shfl_xor(sum, off, 32);
    const float p = ex / sum;

    s_p[wave][lane]   = p;
    s_nbr[wave][lane] = nbr;
    __syncthreads();

    // out[node][h] = sum_j p_j * V[nbr_j][h]; lane handles h=lane and h=lane+32
    float acc0 = 0.0f, acc1 = 0.0f;
    #pragma unroll 4
    for (int j = 0; j < DEG; ++j) {
        const int   nb = s_nbr[wave][j];
        const float pj = s_p[wave][j];
        const float* __restrict__ v = ws + (size_t)nb * WSROW + 2 * NH;
        acc0 += pj * v[lane];
        acc1 += pj * v[lane + 32];
    }
    out[(size_t)node * NH + lane]      = acc0;
    out[(size_t)node * NH + 32 + lane] = acc1;
}

// ---------------------------------------------------------------------------
extern "C" void kernel_launch(void* const* d_in, const int* in_sizes, int n_in,
                              void* d_out, int out_size, void* d_ws, size_t ws_size,
                              hipStream_t stream) {
    const float* x          = (const float*)d_in[0];
    // d_in[1] = adj (dense mask) -- unused: graph structure comes from edge_index
    const int*   edge_index = (const int*)  d_in[2];   // [2, E]
    const int*   edge_type  = (const int*)  d_in[3];   // [E]
    const float* Wq         = (const float*)d_in[4];
    const float* bq         = (const float*)d_in[5];
    const float* Wk         = (const float*)d_in[6];
    const float* bk         = (const float*)d_in[7];
    const float* Wv         = (const float*)d_in[8];
    const float* ekt        = (const float*)d_in[9];   // [16,1]

    float* ws  = (float*)d_ws;     // [8192 x 192] QKV workspace (6.3 MB)
    float* out = (float*)d_out;    // [8192 x 64]
    const int* edge_dst = edge_index + NE;  // second row of edge_index

    // 2048 tiles / 8 waves per block = 256 blocks
    qkv_wmma_kernel<<<256, 256, 0, stream>>>(x, Wq, bq, Wk, bk, Wv, ws);
    // 8192 nodes / 8 waves per block = 1024 blocks
    edge_attn_kernel<<<NNODES / 8, 256, 0, stream>>>(ws, edge_dst, edge_type, ekt, out);
}